// GATModel_20693152432176
// MI455X (gfx1250) — compile-verified
//
#include <hip/hip_runtime.h>
#include <math.h>

// ---------------------------------------------------------------------------
// 3-layer GAT for MI455X (gfx1250, wave32).
//  * GEMMs on the matrix pipe with V_WMMA_F32_16X16X4_F32 (exact fp32 — the
//    problem is L2/HBM-latency bound at ~5 GFLOP total, so bf16 conversion
//    buys nothing and would perturb numerics).
//  * Async GLOBAL_LOAD_ASYNC_TO_LDS_B128 double-buffered tile staging
//    (ASYNCcnt) when the toolchain exposes the builtins; plain-store fallback
//    otherwise.
//  * Deterministic CSR (rank-sorted edges per dst) -> no float atomics,
//    identical summation order every call.
//  * Edge softmax alpha materialized once per (edge,prefix,head).
// Workspace requirement: ~160 MB.
// ---------------------------------------------------------------------------

#define GAT_N_NODES 1500
#define GAT_N_EDGES 12000

typedef __attribute__((ext_vector_type(2))) float v2f;
typedef __attribute__((ext_vector_type(8))) float v8f;
typedef int gat_v4i __attribute__((vector_size(16)));  // b128 payload type

#if defined(__has_builtin)
#if __has_builtin(__builtin_amdgcn_global_load_async_to_lds_b128) && \
    __has_builtin(__builtin_amdgcn_s_wait_asynccnt)
#define GAT_ASYNC 1
#endif
#endif
#ifndef GAT_ASYNC
#define GAT_ASYNC 0
#endif

#if GAT_ASYNC
__device__ __forceinline__ void gat_async_ld128(const void* g, void* l) {
    __builtin_amdgcn_global_load_async_to_lds_b128(
        (__attribute__((address_space(1))) gat_v4i*)g,
        (__attribute__((address_space(3))) gat_v4i*)l, 0, 0);
}
#define GAT_WAIT_ASYNC() __builtin_amdgcn_s_wait_asynccnt(0)
#else
#define GAT_WAIT_ASYNC() ((void)0)
#endif

// ------------------------------ GEMM (WMMA) --------------------------------
// C[M x Nc] = A[M x K] @ B[K x Nc], fp32. Block = 256 thr (8 waves, 2x4),
// block tile 32(M) x 64(N), K double-buffered through LDS in chunks of 16.
// Requires: K % 16 == 0, Nc % 64 == 0.  M is guarded.
//
// LDS pads: As rows 20 floats (80 B: 16B-aligned for b128; 20r mod 64 covers
// each multiple of 4 once over 16 rows -> conflict-free frag reads; upper
// half-wave reads kk+2 -> disjoint banks). Bs rows 72 floats (288 B, 16B
// aligned; half-waves 2 rows apart -> +16 bank offset -> disjoint).
__global__ __launch_bounds__(256) void gat_gemm_wmma(
    const float* __restrict__ A, const float* __restrict__ B,
    float* __restrict__ C, int M, int K, int Nc)
{
    __shared__ float As[2][32][20];
    __shared__ float Bs[2][16][72];

    const int tid   = threadIdx.x;
    const int wave  = tid >> 5;          // 0..7
    const int lane  = tid & 31;
    const int half  = lane >> 4;         // 0/1 (wave32 half)
    const int l15   = lane & 15;         // row (A) / col (B,C/D)
    const int wm    = wave >> 2;         // 0..1 -> 16-row sub-tile
    const int wn    = wave & 3;          // 0..3 -> 16-col sub-tile

    const int mBase = blockIdx.y * 32;
    const int nBase = blockIdx.x * 64;

    // Staging coords (one b128 / 4 floats per issue)
    const int bR = tid >> 4;             // 0..15  (B tile row)
    const int bC = (tid & 15) << 2;      // 0..60  (B tile col)
    const int aR = tid >> 2;             // 0..63  (A tile row; valid < 32)
    const int aC = (tid & 3) << 2;       // 0,4,8,12 (A tile col)

    // Stage one 16-K chunk into buffer `b`.
    auto stage = [&](int b, int k0) {
#if GAT_ASYNC
        gat_async_ld128(&B[(size_t)(k0 + bR) * Nc + nBase + bC], &Bs[b][bR][bC]);
        if (tid < 128) {
            int gr = mBase + aR;
            if (gr < M)   // OOB rows left stale: only feed never-stored C rows
                gat_async_ld128(&A[(size_t)gr * K + k0 + aC], &As[b][aR][aC]);
        }
#else
        for (int i = tid; i < 16 * 64; i += 256) {
            int r = i >> 6, c = i & 63;
            Bs[b][r][c] = B[(size_t)(k0 + r) * Nc + nBase + c];
        }
        for (int i = tid; i < 32 * 16; i += 256) {
            int r = i >> 4, c = i & 15;
            int gr = mBase + r;
            As[b][r][c] = (gr < M) ? A[(size_t)gr * K + k0 + c] : 0.0f;
        }
#endif
    };

    v8f acc = {};
    const int nChunks = K >> 4;

    stage(0, 0);
    GAT_WAIT_ASYNC();
    __syncthreads();

    int cur = 0;
    for (int ch = 0; ch < nChunks; ++ch) {
        int nxt = cur ^ 1;
        if (ch + 1 < nChunks) stage(nxt, (ch + 1) << 4);  // overlap with WMMA

#pragma unroll
        for (int kk = 0; kk < 16; kk += 4) {
            // A 16x4 frag: lanes 0-15 hold K=kk..kk+1, lanes 16-31 K=kk+2..kk+3
            v2f a;
            a.x = As[cur][wm * 16 + l15][kk + 2 * half + 0];
            a.y = As[cur][wm * 16 + l15][kk + 2 * half + 1];
            // B 4x16 frag: VGPR0 row K=kk(+2*half), VGPR1 row K=kk+1(+2*half)
            v2f b;
            b.x = Bs[cur][kk + 2 * half + 0][wn * 16 + l15];
            b.y = Bs[cur][kk + 2 * half + 1][wn * 16 + l15];
            acc = __builtin_amdgcn_wmma_f32_16x16x4_f32(
                false, a, false, b, (short)0, acc, false, false);
        }

        GAT_WAIT_ASYNC();
        __syncthreads();
        cur = nxt;
    }

    // C/D layout: VGPR r -> row (r + 8*half), col l15
#pragma unroll
    for (int r = 0; r < 8; ++r) {
        int row = mBase + wm * 16 + r + 8 * half;
        if (row < M)
            C[(size_t)row * Nc + nBase + wn * 16 + l15] = acc[r];
    }
}

// ------------------------------ CSR build ----------------------------------
__global__ void gat_zero_i32(int* __restrict__ p, int n) {
    int i = blockIdx.x * blockDim.x + threadIdx.x;
    if (i < n) p[i] = 0;
}

__global__ void gat_hist_dst(const int* __restrict__ dst, int* __restrict__ deg) {
    int e = blockIdx.x * blockDim.x + threadIdx.x;
    if (e < GAT_N_EDGES) atomicAdd(&deg[dst[e]], 1);
}

// Exclusive scan of deg[0..N) -> rowptr[0..N]; one block of 256 threads.
__global__ __launch_bounds__(256) void gat_scan_rowptr(
    const int* __restrict__ deg, int* __restrict__ rowptr)
{
    __shared__ int s[256];
    const int CH = (GAT_N_NODES + 255) / 256;  // 6
    int t = threadIdx.x;
    int base = t * CH;
    int sum = 0;
    for (int j = 0; j < CH; ++j) {
        int idx = base + j;
        sum += (idx < GAT_N_NODES) ? deg[idx] : 0;
    }
    s[t] = sum;
    __syncthreads();
    for (int off = 1; off < 256; off <<= 1) {
        int v = (t >= off) ? s[t - off] : 0;
        __syncthreads();
        s[t] += v;
        __syncthreads();
    }
    int run = (t == 0) ? 0 : s[t - 1];
    for (int j = 0; j < CH; ++j) {
        int idx = base + j;
        if (idx < GAT_N_NODES) {
            rowptr[idx] = run;
            run += deg[idx];
        }
    }
    if (t == 255) rowptr[GAT_N_NODES] = s[255];
}

// Deterministic stable fill: edge e goes to slot rowptr[d] + (#earlier edges
// with the same dst). E=12000 -> dst[] is L1-resident, O(E^2) is trivial.
__global__ void gat_rank_fill(const int* __restrict__ dst,
                              const int* __restrict__ rowptr,
                              int* __restrict__ eidx)
{
    int e = blockIdx.x * blockDim.x + threadIdx.x;
    if (e >= GAT_N_EDGES) return;
    int d = dst[e];
    int r = 0;
    for (int j = 0; j < e; ++j) r += (dst[j] == d);
    eidx[rowptr[d] + r] = e;
}

// --------------------------- attention pieces ------------------------------
// el/er[r*H + h] = sum_f feat[r, h, f] * al/ar[h, f]
__global__ void gat_coef(const float* __restrict__ feat,
                         const float* __restrict__ al, const float* __restrict__ ar,
                         float* __restrict__ el, float* __restrict__ er,
                         int R, int H, int Fo)
{
    int i = blockIdx.x * blockDim.x + threadIdx.x;
    if (i >= R * H) return;
    int h = i % H, r = i / H;
    const float* f = feat + (size_t)r * H * Fo + (size_t)h * Fo;
    float sl = 0.f, sr = 0.f;
    for (int j = 0; j < Fo; ++j) {
        float v = f[j];
        sl += v * al[h * Fo + j];
        sr += v * ar[h * Fo + j];
    }
    el[i] = sl;
    er[i] = sr;
}

__device__ __forceinline__ float gat_lrelu(float x) {
    return (x > 0.f) ? x : 0.2f * x;
}

// Per (n, p, h): max & sum of exp over incoming edges (fixed CSR order).
__global__ void gat_softmax_stats(const float* __restrict__ el,
                                  const float* __restrict__ er,
                                  const int* __restrict__ src,
                                  const int* __restrict__ rowptr,
                                  const int* __restrict__ eidx,
                                  float* __restrict__ emax, float* __restrict__ den,
                                  int Pc, int H)
{
    int i = blockIdx.x * blockDim.x + threadIdx.x;
    int total = GAT_N_NODES * Pc * H;
    if (i >= total) return;
    int h = i % H;
    int t = i / H;
    int p = t % Pc;
    int n = t / Pc;
    int b = rowptr[n], e2 = rowptr[n + 1];
    float ern = er[i];  // er[(n*Pc+p)*H + h] == er[i]
    float m = -INFINITY;
    for (int q = b; q < e2; ++q) {
        int s = src[eidx[q]];
        float x = gat_lrelu(el[(s * Pc + p) * H + h] + ern);
        m = fmaxf(m, x);
    }
    float sum = 0.f;
    for (int q = b; q < e2; ++q) {
        int s = src[eidx[q]];
        float x = gat_lrelu(el[(s * Pc + p) * H + h] + ern);
        sum += expf(x - m);
    }
    emax[i] = m;
    den[i] = sum;
}

// alpha[(e*Pc+p)*H + h] = exp(lrelu(el[src]+er[dst]) - emax[dst]) / den[dst]
__global__ void gat_alpha(const float* __restrict__ el, const float* __restrict__ er,
                          const float* __restrict__ emax, const float* __restrict__ den,
                          const int* __restrict__ src, const int* __restrict__ dst,
                          float* __restrict__ alpha, int Pc, int H)
{
    int i = blockIdx.x * blockDim.x + threadIdx.x;
    int total = GAT_N_EDGES * Pc * H;
    if (i >= total) return;
    int h = i % H;
    int t = i / H;
    int p = t % Pc;
    int e = t / Pc;
    int s = src[e], d = dst[e];
    int di = (d * Pc + p) * H + h;
    float x = gat_lrelu(el[(s * Pc + p) * H + h] + er[di]);
    alpha[i] = expf(x - emax[di]) / den[di];
}

// out[r, h, f] = sum_{e in in(n)} alpha[e,p,h] * feat[src(e), p, h, f] + bias[h,f]
// optionally exact GELU.  r = n*Pc + p.
__global__ void gat_aggregate(const float* __restrict__ feat,
                              const float* __restrict__ alpha,
                              const float* __restrict__ bias,
                              const int* __restrict__ rowptr,
                              const int* __restrict__ eidx,
                              const int* __restrict__ src,
                              float* __restrict__ out,
                              int Pc, int H, int Fo, int do_gelu, int total)
{
    int i = blockIdx.x * blockDim.x + threadIdx.x;
    if (i >= total) return;
    int f = i % Fo;
    int t = i / Fo;
    int h = t % H;
    int r = t / H;
    int n = r / Pc;
    int p = r - n * Pc;
    int b = rowptr[n], e2 = rowptr[n + 1];
    float acc = 0.f;
    for (int q = b; q < e2; ++q) {
        int e = eidx[q];
        int s = src[e];
        acc += alpha[(e * Pc + p) * H + h] *
               feat[((size_t)(s * Pc + p) * H + h) * Fo + f];
    }
    float o = acc + bias[h * Fo + f];
    if (do_gelu) o = 0.5f * o * (1.0f + erff(o * 0.70710678118654752440f));
    out[i] = o;
}

// ------------------------------ host side ----------------------------------
static void gat_run_layer(const float* Hin, const float* W, const float* al,
                          const float* ar, const float* bias,
                          float* feat, float* el, float* er, float* emax,
                          float* den, float* alpha, float* out,
                          const int* src, const int* dst,
                          const int* rowptr, const int* eidx,
                          int M, int K, int Nc, int Pc, int H, int Fo,
                          int do_gelu, hipStream_t stream)
{
    dim3 gg(Nc / 64, (M + 31) / 32);
    gat_gemm_wmma<<<gg, 256, 0, stream>>>(Hin, W, feat, M, K, Nc);

    int RH = M * H;
    gat_coef<<<(RH + 255) / 256, 256, 0, stream>>>(feat, al, ar, el, er, M, H, Fo);

    int NPH = GAT_N_NODES * Pc * H;
    gat_softmax_stats<<<(NPH + 255) / 256, 256, 0, stream>>>(
        el, er, src, rowptr, eidx, emax, den, Pc, H);

    int EPH = GAT_N_EDGES * Pc * H;
    gat_alpha<<<(EPH + 255) / 256, 256, 0, stream>>>(
        el, er, emax, den, src, dst, alpha, Pc, H);

    int total = M * H * Fo;
    gat_aggregate<<<(total + 255) / 256, 256, 0, stream>>>(
        feat, alpha, bias, rowptr, eidx, src, out, Pc, H, Fo, do_gelu, total);
}

extern "C" void kernel_launch(void* const* d_in, const int* in_sizes, int n_in,
                              void* d_out, int out_size, void* d_ws, size_t ws_size,
                              hipStream_t stream)
{
    const float* node = (const float*)d_in[0];
    const int*   src  = (const int*)d_in[1];
    const int*   dst  = (const int*)d_in[2];
    const float* W1   = (const float*)d_in[3];
    const float* al1  = (const float*)d_in[4];
    const float* ar1  = (const float*)d_in[5];
    const float* b1   = (const float*)d_in[6];
    const float* W2   = (const float*)d_in[7];
    const float* al2  = (const float*)d_in[8];
    const float* ar2  = (const float*)d_in[9];
    const float* b2   = (const float*)d_in[10];
    const float* W3   = (const float*)d_in[11];
    const float* al3  = (const float*)d_in[12];
    const float* ar3  = (const float*)d_in[13];
    const float* b3   = (const float*)d_in[14];
    float* out = (float*)d_out;

    // Workspace carve-up (worst-case sizes across layers), 256B aligned.
    char* ws = (char*)d_ws;
    size_t off = 0;
    auto carve = [&](size_t bytes) -> void* {
        void* p = ws + off;
        off += (bytes + 255) & ~(size_t)255;
        return p;
    };
    float* feat  = (float*)carve((size_t)24576000 * 4);  // 96000 x 256 max
    float* hbuf  = (float*)carve((size_t)6144000 * 4);   // 12000 x 512 max
    float* alpha = (float*)carve((size_t)6144000 * 4);   // 12000 * 64 * 8 max
    float* el    = (float*)carve((size_t)768000 * 4);    // 96000 * 8 max
    float* er    = (float*)carve((size_t)768000 * 4);
    float* emax  = (float*)carve((size_t)768000 * 4);    // 1500 * 64 * 8 max
    float* den   = (float*)carve((size_t)768000 * 4);
    int* deg     = (int*)carve((size_t)GAT_N_NODES * 4);
    int* rowptr  = (int*)carve((size_t)(GAT_N_NODES + 1) * 4);
    int* eidx    = (int*)carve((size_t)GAT_N_EDGES * 4);
    (void)ws_size; (void)in_sizes; (void)n_in; (void)out_size;

    // --- deterministic CSR over dst (graph is static across layers) ---
    gat_zero_i32<<<(GAT_N_NODES + 255) / 256, 256, 0, stream>>>(deg, GAT_N_NODES);
    gat_hist_dst<<<(GAT_N_EDGES + 255) / 256, 256, 0, stream>>>(dst, deg);
    gat_scan_rowptr<<<1, 256, 0, stream>>>(deg, rowptr);
    gat_rank_fill<<<(GAT_N_EDGES + 255) / 256, 256, 0, stream>>>(dst, rowptr, eidx);

    // --- layer 1: [1500,512] -> [1500,8,64], GELU -> hbuf (1500x512) ---
    gat_run_layer(node, W1, al1, ar1, b1, feat, el, er, emax, den, alpha, hbuf,
                  src, dst, rowptr, eidx,
                  /*M=*/1500, /*K=*/512, /*Nc=*/512, /*Pc=*/1, /*H=*/8, /*Fo=*/64,
                  /*gelu=*/1, stream);

    // --- layer 2: rows = 1500*8, [.,64] -> [.,8,64], GELU -> hbuf (12000x512) ---
    gat_run_layer(hbuf, W2, al2, ar2, b2, feat, el, er, emax, den, alpha, hbuf,
                  src, dst, rowptr, eidx,
                  /*M=*/12000, /*K=*/64, /*Nc=*/512, /*Pc=*/8, /*H=*/8, /*Fo=*/64,
                  /*gelu=*/1, stream);

    // --- layer 3: rows = 1500*64, [.,64] -> [.,8,32] -> d_out (no GELU) ---
    gat_run_layer(hbuf, W3, al3, ar3, b3, feat, el, er, emax, den, alpha, out,
                  src, dst, rowptr, eidx,
                  /*M=*/96000, /*K=*/64, /*Nc=*/256, /*Pc=*/64, /*H=*/8, /*Fo=*/32,
                  /*gelu=*/0, stream);
}